// KnowledgeCircuit_34213709480500
// MI455X (gfx1250) — compile-verified
//
#include <hip/hip_runtime.h>

// Problem constants from the reference: B=4, S=2048, D=1024, N=65536, K=16
#define TOKENS   (4 * 2048)
#define DIM      1024
#define NEXPERT  16
#define BLOCK    256

typedef unsigned int v4u __attribute__((ext_vector_type(4)));
typedef unsigned int v8u __attribute__((ext_vector_type(8)));

// Low 32 bits of a flat shared-aperture pointer are the wave-relative LDS byte
// address (CDNA5 aperture rule).
__device__ __forceinline__ unsigned lds_byte_addr(const void* p) {
    return (unsigned)(unsigned long long)p;
}

// Issue one TDM gather: 16 rows (selected by 16-bit indices) of a row-major
// [65536 x 1024] f32 tensor -> contiguous 64 KB LDS tile.
// D# layout per CDNA5 ISA (gather mode).
__device__ __forceinline__ void tdm_gather16_rows(const float* base,
                                                  unsigned lds_byte,
                                                  v4u idx_lo, v4u idx_hi)
{
    const unsigned long long ga = (unsigned long long)base;

    v4u g0;
    g0.x = 1u | (1u << 31);                 // count=1, gather_mode=1, 16-bit idx
    g0.y = lds_byte;                        // lds_addr (bytes)
    g0.z = (unsigned)(ga & 0xffffffffu);    // global_addr[31:0]
    g0.w = (unsigned)((ga >> 32) & 0x01ffffffu) | (2u << 30); // addr[56:32], type=2

    v8u g1;
    g1[0] = (2u << 16);                     // workgroup_mask=0, data_size=2 (4B)
    g1[1] = (unsigned)(DIM & 0xffff) << 16; // tensor_dim0[15:0]=1024
    g1[2] = 0u;                             // tensor_dim0[31:16]=0, tensor_dim1[15:0]=0
    g1[3] = 1u | ((unsigned)DIM << 16);     // tensor_dim1[31:16]=1 (65536), tile_dim0=1024
    g1[4] = (unsigned)NEXPERT;              // tile_dim1 = #valid indices = 16
    g1[5] = (unsigned)DIM;                  // tensor_dim0_stride[31:0] = 1024
    g1[6] = 0u;                             // stride0 hi, stride1 lo (ignored in gather)
    g1[7] = 0u;

    asm volatile("tensor_load_to_lds %0, %1, %2, %3"
                 :
                 : "s"(g0), "s"(g1), "s"(idx_lo), "s"(idx_hi)
                 : "memory");
}

__global__ __launch_bounds__(BLOCK)
void kc_gather_sense_emit(const float* __restrict__ x,
                          const float* __restrict__ emb,
                          const float* __restrict__ w,
                          const float* __restrict__ gate,
                          const int*   __restrict__ topk,
                          float*       __restrict__ out)
{
    extern __shared__ float smem[];         // dynamic: [0,16K) emb tile, [16K,32K) w tile
    float* __restrict__ s_emb = smem;                   // 16 rows x 1024 f32 (64 KB)
    float* __restrict__ s_w   = smem + NEXPERT * DIM;   // 16 rows x 1024 f32 (64 KB)

    __shared__ float s_x[DIM];              // staged x row (4 KB)
    __shared__ float s_g[NEXPERT];          // raw gate values
    __shared__ float s_act[NEXPERT];        // gated activations (sense result)

    const int tok  = blockIdx.x;            // 0 .. 8191
    const int tid  = threadIdx.x;           // 0 .. 255
    const int lane = tid & 31;
    const int wave = tid >> 5;              // 0 .. 7

    // ---- Async stage of the x row into LDS (all 8 waves, 16 B per lane) ----
    {
        unsigned lds_dst = lds_byte_addr(&s_x[tid * 4]);
        unsigned goff    = (unsigned)(tid * 16);
        unsigned long long gbase =
            (unsigned long long)(x + (size_t)tok * DIM);
        asm volatile(
            "global_load_async_to_lds_b128 %0, %1, %2 offset:0"
            :
            : "v"(lds_dst), "v"(goff), "s"(gbase)
            : "memory");
    }

    // ---- Wave 0: build the gather descriptor and launch both TDM DMAs ----
    if (wave == 0) {
        int my = 0;
        if (lane < NEXPERT) my = topk[(size_t)tok * NEXPERT + lane];
        #define RL(j) ((unsigned)__builtin_amdgcn_readlane(my, (j)) & 0xffffu)
        v4u lo, hi;
        lo.x = RL(0)  | (RL(1)  << 16);
        lo.y = RL(2)  | (RL(3)  << 16);
        lo.z = RL(4)  | (RL(5)  << 16);
        lo.w = RL(6)  | (RL(7)  << 16);
        hi.x = RL(8)  | (RL(9)  << 16);
        hi.y = RL(10) | (RL(11) << 16);
        hi.z = RL(12) | (RL(13) << 16);
        hi.w = RL(14) | (RL(15) << 16);
        #undef RL

        tdm_gather16_rows(emb, lds_byte_addr(s_emb), lo, hi);  // needed for sense
        tdm_gather16_rows(w,   lds_byte_addr(s_w),   lo, hi);  // needed for emit

        // Tensor ops from one wave complete in order: cnt<=1 => emb tile done.
        asm volatile("s_wait_tensorcnt 0x1" ::: "memory");
    }

    // Stage the gates (uniform per token).
    if (tid < NEXPERT) {
        s_g[tid] = gate[(size_t)tok * NEXPERT + tid];
    }

    // Own-wave async x transfer done, then cross-wave visibility.
    asm volatile("s_wait_asynccnt 0" ::: "memory");
    __syncthreads();   // s_x, s_g and the emb tile are now valid

    // ---- Phase 1: sense. Each wave computes 2 of the 16 dot products, ----
    // ---- reading both operands from LDS (w-tile DMA still in flight).  ----
    {
        const int e0 = 2 * wave;
        const int e1 = 2 * wave + 1;
        const float* __restrict__ r0 = s_emb + e0 * DIM;
        const float* __restrict__ r1 = s_emb + e1 * DIM;

        float a0 = 0.0f, a1 = 0.0f;
        #pragma unroll
        for (int i = 0; i < DIM / 4; i += 32) {        // 8 iterations
            const int q = i + lane;                    // float4 index
            const float4 xv  = *(const float4*)&s_x[q * 4];   // ds b128
            const float4 e0v = *(const float4*)&r0[q * 4];    // ds b128
            const float4 e1v = *(const float4*)&r1[q * 4];    // ds b128
            a0 = fmaf(xv.x, e0v.x, a0); a0 = fmaf(xv.y, e0v.y, a0);
            a0 = fmaf(xv.z, e0v.z, a0); a0 = fmaf(xv.w, e0v.w, a0);
            a1 = fmaf(xv.x, e1v.x, a1); a1 = fmaf(xv.y, e1v.y, a1);
            a1 = fmaf(xv.z, e1v.z, a1); a1 = fmaf(xv.w, e1v.w, a1);
        }
        // wave32 butterfly reduction
        #pragma unroll
        for (int off = 16; off >= 1; off >>= 1) {
            a0 += __shfl_xor(a0, off, 32);
            a1 += __shfl_xor(a1, off, 32);
        }
        if (lane == 0) {
            s_act[e0] = a0 * s_g[e0];
            s_act[e1] = a1 * s_g[e1];
        }
    }

    // w tile must be fully landed before the emit phase reads it.
    if (wave == 0) {
        asm volatile("s_wait_tensorcnt 0x0" ::: "memory");
    }
    __syncthreads();   // s_act and the w tile are now valid

    // ---- Phase 2: emit. Thread t owns output elements 4t..4t+3, all from LDS. ----
    {
        const int dbase = tid * 4;
        float4 acc = make_float4(0.f, 0.f, 0.f, 0.f);
        #pragma unroll
        for (int k = 0; k < NEXPERT; ++k) {
            const float g = s_act[k];                          // LDS broadcast
            const float4 wv = *(const float4*)&s_w[k * DIM + dbase]; // ds b128
            acc.x = fmaf(g, wv.x, acc.x);
            acc.y = fmaf(g, wv.y, acc.y);
            acc.z = fmaf(g, wv.z, acc.z);
            acc.w = fmaf(g, wv.w, acc.w);
        }
        *(float4*)&out[(size_t)tok * DIM + dbase] = acc;       // coalesced b128
    }
}

extern "C" void kernel_launch(void* const* d_in, const int* in_sizes, int n_in,
                              void* d_out, int out_size, void* d_ws, size_t ws_size,
                              hipStream_t stream) {
    const float* x    = (const float*)d_in[0];  // [B,S,D]
    const float* emb  = (const float*)d_in[1];  // [N,D]
    const float* w    = (const float*)d_in[2];  // [N,D]
    const float* gate = (const float*)d_in[3];  // [B,S,K]
    const int*   idx  = (const int*)  d_in[4];  // [B,S,K]
    float*       out  = (float*)d_out;          // [B,S,D]

    (void)in_sizes; (void)n_in; (void)out_size; (void)d_ws; (void)ws_size;

    const size_t dyn_lds = 2u * NEXPERT * DIM * sizeof(float);  // 128 KB tiles
    hipFuncSetAttribute(reinterpret_cast<const void*>(kc_gather_sense_emit),
                        hipFuncAttributeMaxDynamicSharedMemorySize,
                        (int)dyn_lds);

    kc_gather_sense_emit<<<dim3(TOKENS), dim3(BLOCK), dyn_lds, stream>>>(
        x, emb, w, gate, idx, out);
}